// TrainSingleState_90915867722239
// MI455X (gfx1250) — compile-verified
//
#include <hip/hip_runtime.h>

#define B_    8192
#define T_    256
#define F_    18
#define U_    32
#define NIDS  1000
#define NWG   64
#define NTHR  256
#define TF    (T_ * F_)

typedef __attribute__((ext_vector_type(2))) float v2f;
typedef __attribute__((ext_vector_type(8))) float v8f;

// D = A(16x4 f32) * B(4x16 f32) + C(16x16 f32)   -> v_wmma_f32_16x16x4_f32
__device__ __forceinline__ v8f wmma4(v2f a, v2f b, v8f c) {
  return __builtin_amdgcn_wmma_f32_16x16x4_f32(false, a, false, b, (short)0, c,
                                               false, false);
}

#if __has_builtin(__builtin_amdgcn_tanhf)
__device__ __forceinline__ float fast_tanh(float x) {
  return __builtin_amdgcn_tanhf(x);  // v_tanh_f32 (gfx1250 TRANS op)
}
#else
__device__ __forceinline__ float fast_tanh(float x) { return tanhf(x); }
#endif

__device__ __forceinline__ float fast_sigmoid(float x) {
#if __has_builtin(__builtin_amdgcn_exp2f) && __has_builtin(__builtin_amdgcn_rcpf)
  return __builtin_amdgcn_rcpf(1.0f +
                               __builtin_amdgcn_exp2f(-1.442695041f * x));
#else
  return 1.0f / (1.0f + expf(-x));
#endif
}

// Monotonic device-wide barrier: each call i expects counter >= NWG*(i+1).
__device__ __forceinline__ void grid_barrier(unsigned* cnt, unsigned expected) {
  __syncthreads();
  if (threadIdx.x == 0) {
    __threadfence();  // release prior table stores to device scope
    __hip_atomic_fetch_add(cnt, 1u, __ATOMIC_ACQ_REL, __HIP_MEMORY_SCOPE_AGENT);
    while (__hip_atomic_load(cnt, __ATOMIC_ACQUIRE, __HIP_MEMORY_SCOPE_AGENT) <
           expected) {
      __builtin_amdgcn_s_sleep(1);
    }
  }
  __syncthreads();
}

__global__ void init_state(const float* __restrict__ sinit,
                           float* __restrict__ table,
                           unsigned* __restrict__ bar, int n) {
  int i = blockIdx.x * blockDim.x + threadIdx.x;
  if (i < n) table[i] = sinit[i];
  if (i == 0) *bar = 0u;
}

__global__ __launch_bounds__(NTHR) void gru_persistent(
    const float* __restrict__ x, const float* __restrict__ g_kern,
    const float* __restrict__ g_rec, const float* __restrict__ g_bias,
    const float* __restrict__ g_w1, const float* __restrict__ g_b1,
    const float* __restrict__ g_w2, const float* __restrict__ g_b2,
    float* __restrict__ table, unsigned* __restrict__ bar,
    float* __restrict__ out) {
  // Weights pre-swizzled into WMMA B-fragment order:
  //   pair index = (kstep * 2 + laneHalf) * ncols + col
  //   value      = { W[4ks + 2*half][col], W[4ks + 2*half + 1][col] }
  // x-kernel K padded 18->20; virtual row 18 holds the bias (A feeds 1.0),
  // row 19 is zero.  So xk = x@kernel + bias comes out of the matrix pipe.
  __shared__ v2f s_kernp[5 * 2 * 96];   // (7680 B)
  __shared__ v2f s_recp[8 * 2 * 96];    // [Uz | Ur | Uh] (12288 B)
  __shared__ v2f s_w1p[8 * 2 * 32];     // head w1 (4096 B)
  __shared__ float s_b1[32];
  __shared__ float s_w2[32];
  __shared__ float s_b2[1];
  __shared__ float s_h[8][16 * 32];     // per-wave h tile (row-major 16x32)
  __shared__ float s_rh[8][16 * 32];    // per-wave r*h tile
  __shared__ int s_id[8][16];           // ids pre-scaled by U

  const int tid = threadIdx.x;
  const int lane = tid & 31;
  const int wave = tid >> 5;
  const int c16 = lane & 15;
  const int koff = (lane < 16) ? 0 : 2;    // A-frag K offset for upper half
  const int rbase = (lane < 16) ? 0 : 8;   // C/D row base for upper half
  const int hb96 = (lane < 16) ? 0 : 96;   // B-frag half select (96-col mats)
  const int hb32 = (lane < 16) ? 0 : 32;   // B-frag half select (32-col mats)
  const int r0 = (blockIdx.x * 8 + wave) * 16;  // 16 batch rows per wave

  // ---- stage weights into LDS (swizzled) ----
  for (int i = tid; i < 5 * 2 * 96; i += NTHR) {
    int ks = i / 192, rem = i - ks * 192;
    int half = rem / 96, c = rem - half * 96;
    int k0 = ks * 4 + half * 2;
    v2f v;
    v.x = (k0 < F_) ? g_kern[k0 * 96 + c] : (k0 == F_ ? g_bias[c] : 0.0f);
    v.y = (k0 + 1 < F_) ? g_kern[(k0 + 1) * 96 + c]
                        : (k0 + 1 == F_ ? g_bias[c] : 0.0f);
    s_kernp[i] = v;
  }
  for (int i = tid; i < 8 * 2 * 96; i += NTHR) {
    int ks = i / 192, rem = i - ks * 192;
    int half = rem / 96, c = rem - half * 96;
    int k0 = ks * 4 + half * 2;
    v2f v;
    v.x = g_rec[k0 * 96 + c];
    v.y = g_rec[(k0 + 1) * 96 + c];
    s_recp[i] = v;
  }
  for (int i = tid; i < 8 * 2 * 32; i += NTHR) {
    int ks = i / 64, rem = i - ks * 64;
    int half = rem / 32, c = rem - half * 32;
    int k0 = ks * 4 + half * 2;
    v2f v;
    v.x = g_w1[k0 * 32 + c];
    v.y = g_w1[(k0 + 1) * 32 + c];
    s_w1p[i] = v;
  }
  for (int i = tid; i < 32; i += NTHR) s_b1[i] = g_b1[i];
  for (int i = tid; i < 32; i += NTHR) s_w2[i] = g_w2[i];
  if (tid == 0) s_b2[0] = g_b2[0];
  __syncthreads();

  float* hbuf = &s_h[wave][0];
  float* rhbuf = &s_rh[wave][0];
  unsigned bexp = 0;

#pragma unroll 1
  for (int t = 0; t < T_; ++t) {
    // ---- ids + gather h = table[id] (state after step t-1) ----
    if (lane < 16)
      s_id[wave][lane] = (int)x[(size_t)(r0 + lane) * TF + t * F_] * U_;
    __syncthreads();
#pragma unroll
    for (int i = 0; i < 16; ++i)
      hbuf[i * 32 + lane] = table[s_id[wave][i] + lane];
    __syncthreads();

    grid_barrier(bar, (bexp += NWG));  // all gathers before any scatter

    // ---- xk = x_t @ [kernel; bias] : 6 col-tiles of 16, C starts inline-0 ----
    v8f acc[6];
    const float* xrow = x + (size_t)(r0 + c16) * TF + t * F_;
    {  // ks = 0, accumulator seeded with inline 0
      v2f a = *(const v2f*)(xrow + koff);
      const v8f cz = {};
#pragma unroll
      for (int n = 0; n < 6; ++n)
        acc[n] = wmma4(a, s_kernp[hb96 + n * 16 + c16], cz);
    }
#pragma unroll
    for (int ks = 1; ks < 5; ++ks) {
      int kk = ks * 4 + koff;  // even; kk < 18 implies kk+1 < 18
      v2f a;
      if (kk < F_) {
        a = *(const v2f*)(xrow + kk);  // one global_load_b64
      } else {
        a.x = 1.0f;  // virtual K=18 input feeds the bias row
        a.y = 0.0f;
      }
#pragma unroll
      for (int n = 0; n < 6; ++n) {
        v2f b = s_kernp[ks * 192 + hb96 + n * 16 + c16];  // one ds_load_b64
        acc[n] = wmma4(a, b, acc[n]);
      }
    }

    // ---- z,r pre-act += h @ [Uz|Ur] : 4 col-tiles ----
#pragma unroll
    for (int ks = 0; ks < 8; ++ks) {
      int kk = ks * 4 + koff;
      v2f a = *(const v2f*)(hbuf + c16 * 32 + kk);  // aligned ds_load_b64
#pragma unroll
      for (int n = 0; n < 4; ++n) {
        v2f b = s_recp[ks * 192 + hb96 + n * 16 + c16];
        acc[n] = wmma4(a, b, acc[n]);
      }
    }

    // ---- activations, r*h (C/D layout), stash rh for A-frag reads ----
    v8f hcd[2];
#pragma unroll
    for (int n = 0; n < 2; ++n) {
#pragma unroll
      for (int v = 0; v < 8; ++v) {
        float z = fminf(fmaxf(0.2f * acc[n][v] + 0.5f, 0.0f), 1.0f);
        float r = fminf(fmaxf(0.2f * acc[n + 2][v] + 0.5f, 0.0f), 1.0f);
        float h = hbuf[(rbase + v) * 32 + n * 16 + c16];
        hcd[n][v] = h;
        acc[n][v] = z;  // keep z
        rhbuf[(rbase + v) * 32 + n * 16 + c16] = r * h;
      }
    }
    __syncthreads();

    // ---- hh pre-act += (r*h) @ Uh : 2 col-tiles ----
#pragma unroll
    for (int ks = 0; ks < 8; ++ks) {
      int kk = ks * 4 + koff;
      v2f a = *(const v2f*)(rhbuf + c16 * 32 + kk);
#pragma unroll
      for (int n = 0; n < 2; ++n) {
        v2f b = s_recp[ks * 192 + hb96 + 64 + n * 16 + c16];
        acc[4 + n] = wmma4(a, b, acc[4 + n]);
      }
    }

    // ---- h_new = z*h + (1-z)*tanh(hh); overwrite h tile ----
#pragma unroll
    for (int n = 0; n < 2; ++n) {
#pragma unroll
      for (int v = 0; v < 8; ++v) {
        float z = acc[n][v];
        float hh = fast_tanh(acc[4 + n][v]);
        hbuf[(rbase + v) * 32 + n * 16 + c16] =
            z * hcd[n][v] + (1.0f - z) * hh;
      }
    }
    __syncthreads();

    // ---- scatter h_new into the shared table ----
#pragma unroll
    for (int i = 0; i < 16; ++i)
      table[s_id[wave][i] + lane] = hbuf[i * 32 + lane];

    if (t + 1 < T_ && lane < 16)
      __builtin_prefetch(&x[(size_t)(r0 + lane) * TF + (t + 1) * F_], 0, 1);

    grid_barrier(bar, (bexp += NWG));  // scatters before next gather
  }

  // ---- head: hid = relu(h_last @ w1 + b1) via WMMA ----
  v8f hacc[2];
#pragma unroll
  for (int n = 0; n < 2; ++n) {
    float bv = s_b1[n * 16 + c16];
#pragma unroll
    for (int v = 0; v < 8; ++v) hacc[n][v] = bv;
  }
#pragma unroll
  for (int ks = 0; ks < 8; ++ks) {
    int kk = ks * 4 + koff;
    v2f a = *(const v2f*)(hbuf + c16 * 32 + kk);
#pragma unroll
    for (int n = 0; n < 2; ++n) {
      v2f b = s_w1p[ks * 64 + hb32 + n * 16 + c16];
      hacc[n] = wmma4(a, b, hacc[n]);
    }
  }
#pragma unroll
  for (int n = 0; n < 2; ++n)
#pragma unroll
    for (int v = 0; v < 8; ++v)
      rhbuf[(rbase + v) * 32 + n * 16 + c16] = fmaxf(hacc[n][v], 0.0f);
  __syncthreads();

  // out = sigmoid(hid @ w2 + b2): N==1, do it as a per-row dot product
  if (lane < 16) {
    float s = s_b2[0];
#pragma unroll
    for (int k = 0; k < 32; ++k) s += rhbuf[lane * 32 + k] * s_w2[k];
    out[r0 + lane] = fast_sigmoid(s);
  }
}

extern "C" void kernel_launch(void* const* d_in, const int* in_sizes, int n_in,
                              void* d_out, int out_size, void* d_ws,
                              size_t ws_size, hipStream_t stream) {
  (void)in_sizes; (void)n_in; (void)out_size; (void)ws_size;
  const float* x     = (const float*)d_in[0];
  const float* kern  = (const float*)d_in[1];
  const float* rec   = (const float*)d_in[2];
  const float* bias  = (const float*)d_in[3];
  const float* sinit = (const float*)d_in[4];
  const float* w1    = (const float*)d_in[5];
  const float* b1    = (const float*)d_in[6];
  const float* w2    = (const float*)d_in[7];
  const float* b2    = (const float*)d_in[8];

  float* table  = (float*)d_ws;                       // 1000*32 f32 = 128 KB
  unsigned* bar = (unsigned*)((char*)d_ws + 131072);  // barrier counter

  int n = NIDS * U_;
  init_state<<<(n + 255) / 256, 256, 0, stream>>>(sinit, table, bar, n);
  gru_persistent<<<NWG, NTHR, 0, stream>>>(x, kern, rec, bias, w1, b1, w2, b2,
                                           table, bar, (float*)d_out);
}